// LIF_52707838656665
// MI455X (gfx1250) — compile-verified
//
#include <hip/hip_runtime.h>
#include <cstdint>

// LIF scan: y[t] = H(0.5*v + x[t] - 1), v' = fire ? 0 : m.
// Shapes fixed by reference: T=32, B*N = 2,097,152 (divisible by 1024).
// Memory-bound (512 MB @ 23.3 TB/s ~ 22 us). Uses CDNA5 async global->LDS
// pipeline (ASYNCcnt) to keep 8 x 128b loads in flight per lane.

#define T_STEPS 32
#define DEPTH   8                     // async pipeline depth (timesteps)
#define TPB     256
#define EPT     4                     // floats per thread per timestep
#define TILE_ELEMS (TPB * EPT)        // 1024 floats per timestep tile
#define TILE_BYTES (TILE_ELEMS * 4)   // 4 KB per buffer slot

typedef float f32x4 __attribute__((ext_vector_type(4)));

__global__ __launch_bounds__(TPB) void lif_scan_kernel(
    const float* __restrict__ x, float* __restrict__ y, unsigned BN) {
  // 8-deep circular staging buffer: 32 KB (WGP has 320 KB LDS).
  __shared__ __align__(16) float tile[DEPTH * TILE_ELEMS];

  const unsigned tid     = threadIdx.x;
  const unsigned i0      = blockIdx.x * TILE_ELEMS + tid * EPT; // elem idx in (B*N)
  const unsigned xbyte0  = i0 * 4u;                 // byte offset at t=0 (<= 8 MB)
  const unsigned stepB   = BN * 4u;                 // 8 MB stride between timesteps
  // Low 32 bits of the flat address of an LDS object == wave-relative LDS
  // byte address (flat->LDS mapping truncates the upper bits).
  const unsigned lds0    = (unsigned)(uintptr_t)tile + tid * 16u;

  // Per-lane 16B async copy, GVS mode: s[base] + v[32-bit byte offset].
  // s_wait_dscnt 0 first: the slot being overwritten was just ds_load'ed;
  // make sure that read has drained before the async engine rewrites LDS.
#define LIF_PREFETCH(TT, SLOT)                                               \
  do {                                                                       \
    unsigned voff_  = xbyte0 + (unsigned)(TT) * stepB;                       \
    unsigned laddr_ = lds0 + (unsigned)(SLOT) * TILE_BYTES;                  \
    asm volatile("s_wait_dscnt 0\n\t"                                        \
                 "global_load_async_to_lds_b128 %0, %1, %2"                  \
                 :: "v"(laddr_), "v"(voff_), "s"(x)                          \
                 : "memory");                                                \
  } while (0)

  // Prologue: fill the pipeline with timesteps 0..DEPTH-1.
#pragma unroll
  for (int t = 0; t < DEPTH; ++t) LIF_PREFETCH(t, t);

  f32x4 v = {0.f, 0.f, 0.f, 0.f};

  // Fully unrolled so every wait immediate / slot index is a literal.
#pragma unroll
  for (int t = 0; t < T_STEPS; ++t) {
    if (t < T_STEPS - DEPTH) {
      // 8 in-order async loads outstanding; <=7 left => load for t is done.
      asm volatile("s_wait_asynccnt 7" ::: "memory");
    } else if (t == T_STEPS - DEPTH) {
      // No more prefetches issued: drain everything once, then free-run.
      asm volatile("s_wait_asynccnt 0" ::: "memory");
    }

    const int slot = t % DEPTH;  // compile-time constant under full unroll
    const f32x4 xv =
        *(const f32x4*)(tile + slot * TILE_ELEMS + tid * EPT);  // ds_load_b128

    f32x4 yv;
#pragma unroll
    for (int e = 0; e < EPT; ++e) {
      float m   = 0.5f * v[e] + xv[e];   // tau * v + x
      bool fire = (m >= 1.0f);           // H(m - v_th), v_th = 1
      yv[e]     = fire ? 1.0f : 0.0f;
      v[e]      = fire ? 0.0f : m;       // hard reset
    }

    // Streaming output, never re-read: non-temporal 128b store.
    __builtin_nontemporal_store(yv, (f32x4*)(y + (size_t)t * BN + i0));

    if (t < T_STEPS - DEPTH) LIF_PREFETCH(t + DEPTH, slot);
  }
#undef LIF_PREFETCH
}

extern "C" void kernel_launch(void* const* d_in, const int* in_sizes, int n_in,
                              void* d_out, int out_size, void* d_ws, size_t ws_size,
                              hipStream_t stream) {
  (void)n_in; (void)out_size; (void)d_ws; (void)ws_size;
  const float* x = (const float*)d_in[0];
  float*       y = (float*)d_out;

  const unsigned total = (unsigned)in_sizes[0];      // T*B*N = 67,108,864
  const unsigned BN    = total / T_STEPS;            // 2,097,152
  const unsigned grid  = BN / TILE_ELEMS;            // 2048 (exact)

  lif_scan_kernel<<<grid, TPB, 0, stream>>>(x, y, BN);
}